// CGCNN_3496103379077
// MI455X (gfx1250) — compile-verified
//
#include <hip/hip_runtime.h>
#include <hip/hip_bf16.h>
#include <math.h>

// CGCNN on MI455X (gfx1250). f32 WMMA 16x16x4 for all GEMMs (exact precision),
// batchnorm via two-pass recompute (avoids materializing 819MB [E,128] tensors),
// gathers/scatters hit the 192MB L2 (h_src/h_dst = 51MB each, h = 25.6MB).

#define NF   64
#define NF2  128
#define EFD  40
#define LAYERS 3
#define AFEAT 200

typedef __attribute__((ext_vector_type(2))) float v2f;
typedef __attribute__((ext_vector_type(8))) float v8f;

static __device__ __forceinline__ v8f wmma4(v2f a, v2f b, v8f c) {
  // V_WMMA_F32_16X16X4_F32 : D = A(16x4,f32) * B(4x16,f32) + C(16x16,f32)
  return __builtin_amdgcn_wmma_f32_16x16x4_f32(false, a, false, b, (short)0, c,
                                               false, false);
}

static __device__ __forceinline__ v8f zero8() {
  v8f z;
  #pragma unroll
  for (int i = 0; i < 8; ++i) z[i] = 0.0f;
  return z;
}

static __device__ __forceinline__ float sigmoid_f(float v) {
  return 1.0f / (1.0f + __expf(-v));
}
static __device__ __forceinline__ float softplus_f(float v) {
  return (v > 20.0f) ? v : log1pf(__expf(v));
}
// RBF: centers = linspace(0,8,40), step = 8/39, gamma = 39/8 = 4.875
static __device__ __forceinline__ float rbf_f(float d, int k) {
  float t = d - (8.0f / 39.0f) * (float)k;
  return __expf(-4.875f * t * t);
}

// ---------------------------------------------------------------- embedding
__global__ void embed_kernel(const int* __restrict__ atypes,
                             const float* __restrict__ table,
                             const float* __restrict__ W,
                             const float* __restrict__ b,
                             float* __restrict__ x, int N_) {
  int idx = blockIdx.x * blockDim.x + threadIdx.x;
  int total = N_ * NF;
  for (; idx < total; idx += gridDim.x * blockDim.x) {
    int node = idx >> 6, f = idx & 63;
    const float* row = table + atypes[node] * AFEAT;  // 76KB table: cache-hot
    float acc = b[f];
    #pragma unroll 8
    for (int k = 0; k < AFEAT; ++k) acc += row[k] * W[k * NF + f];
    x[idx] = acc;
  }
}

// ------------------------------------------- node projection: out = x@W + b
__global__ void __launch_bounds__(256)
node_proj_kernel(const float* __restrict__ x, const float* __restrict__ W,
                 const float* __restrict__ bias, float* __restrict__ out, int N_) {
  __shared__ float Wl[NF * NF2];  // 32KB
  for (int i = threadIdx.x; i < NF * NF2; i += blockDim.x) Wl[i] = W[i];
  __syncthreads();
  const int lane = threadIdx.x & 31, wave = threadIdx.x >> 5;
  const int col16 = lane & 15, half = lane >> 4;
  const int nwaves = (blockDim.x >> 5) * gridDim.x;
  const int wid = blockIdx.x * (blockDim.x >> 5) + wave;
  const int ntiles = N_ >> 4;  // N divisible by 16
  for (int tile = wid; tile < ntiles; tile += nwaves) {
    const int base = tile << 4;
    const int arow = base + col16;
    v8f acc[8];
    #pragma unroll
    for (int t = 0; t < 8; ++t) acc[t] = zero8();
    #pragma unroll
    for (int kk = 0; kk < NF / 4; ++kk) {
      const int k0 = kk * 4 + 2 * half;
      v2f a;
      a.x = x[arow * NF + k0];
      a.y = x[arow * NF + k0 + 1];
      #pragma unroll
      for (int t = 0; t < 8; ++t) {
        v2f bf;
        bf.x = Wl[k0 * NF2 + t * 16 + col16];
        bf.y = Wl[(k0 + 1) * NF2 + t * 16 + col16];
        acc[t] = wmma4(a, bf, acc[t]);
      }
    }
    #pragma unroll
    for (int t = 0; t < 8; ++t) {
      const int col = t * 16 + col16;
      const float bv = bias[col];
      #pragma unroll
      for (int v = 0; v < 8; ++v)
        out[(size_t)(base + v + 8 * half) * NF2 + col] = acc[t][v] + bv;
    }
  }
}

// ---- edge passes: m = rbf(d)@We + be + hsrc[src] + hdst[dst]
// STATS=1: accumulate per-channel sum / sum-of-squares of m
// STATS=0: normalize (msg_a*m + msg_b), gate sigmoid*softplus, scatter to h
template <int STATS>
__global__ void __launch_bounds__(256)
edge_kernel(const float* __restrict__ dist, const int* __restrict__ src,
            const int* __restrict__ dst, const float* __restrict__ We,
            const float* __restrict__ be, const float* __restrict__ hsrc,
            const float* __restrict__ hdst, float* __restrict__ msg_sum,
            float* __restrict__ msg_ssq, const float* __restrict__ msg_a,
            const float* __restrict__ msg_b, float* __restrict__ hout, int E_) {
  __shared__ float Wl[EFD * NF2];  // 20KB
  __shared__ float s_sum[NF2], s_ssq[NF2];
  for (int i = threadIdx.x; i < EFD * NF2; i += blockDim.x) Wl[i] = We[i];
  if (STATS)
    for (int i = threadIdx.x; i < NF2; i += blockDim.x) { s_sum[i] = 0.f; s_ssq[i] = 0.f; }
  __syncthreads();

  const int lane = threadIdx.x & 31, wave = threadIdx.x >> 5;
  const int col16 = lane & 15, half = lane >> 4;
  const int nwaves = (blockDim.x >> 5) * gridDim.x;
  const int wid = blockIdx.x * (blockDim.x >> 5) + wave;
  const int ntiles = E_ >> 4;  // E divisible by 16

  float ps[8], pss[8], ma[8], mb[8], bev[8];
  #pragma unroll
  for (int t = 0; t < 8; ++t) {
    ps[t] = 0.f; pss[t] = 0.f;
    const int col = t * 16 + col16;
    bev[t] = be[col];
    if (!STATS) { ma[t] = msg_a[col]; mb[t] = msg_b[col]; }
  }

  for (int tile = wid; tile < ntiles; tile += nwaves) {
    const int base = tile << 4;
    const float d = dist[base + col16];
    int se[8], de[8];
    #pragma unroll
    for (int v = 0; v < 8; ++v) {
      const int r = base + v + 8 * half;
      se[v] = src[r];
      de[v] = dst[r];
    }
    v8f acc[8];
    #pragma unroll
    for (int t = 0; t < 8; ++t) acc[t] = zero8();
    #pragma unroll
    for (int kk = 0; kk < EFD / 4; ++kk) {
      const int k0 = kk * 4 + 2 * half;
      v2f a;
      a.x = rbf_f(d, k0);
      a.y = rbf_f(d, k0 + 1);
      #pragma unroll
      for (int t = 0; t < 8; ++t) {
        v2f bf;
        bf.x = Wl[k0 * NF2 + t * 16 + col16];
        bf.y = Wl[(k0 + 1) * NF2 + t * 16 + col16];
        acc[t] = wmma4(a, bf, acc[t]);
      }
    }
    // bias + gathered node projections (L2-resident tables)
    #pragma unroll
    for (int t = 0; t < 8; ++t) {
      const int col = t * 16 + col16;
      #pragma unroll
      for (int v = 0; v < 8; ++v) {
        float m = acc[t][v] + bev[t] + hsrc[(size_t)se[v] * NF2 + col] +
                  hdst[(size_t)de[v] * NF2 + col];
        acc[t][v] = m;
        if (STATS) { ps[t] += m; pss[t] += m * m; }
      }
    }
    if (!STATS) {
      // col c (tiles 0..3) pairs with col c+64 (tiles 4..7) on the same lane
      #pragma unroll
      for (int t = 0; t < 4; ++t) {
        #pragma unroll
        for (int v = 0; v < 8; ++v) {
          const float f = ma[t] * acc[t][v] + mb[t];
          const float s = ma[t + 4] * acc[t + 4][v] + mb[t + 4];
          const float gate = sigmoid_f(f) * softplus_f(s);
          atomicAdd(hout + (size_t)de[v] * NF + t * 16 + col16, gate);
        }
      }
    }
  }
  if (STATS) {
    #pragma unroll
    for (int t = 0; t < 8; ++t) {
      atomicAdd(&s_sum[t * 16 + col16], ps[t]);
      atomicAdd(&s_ssq[t * 16 + col16], pss[t]);
    }
    __syncthreads();
    for (int i = threadIdx.x; i < NF2; i += blockDim.x) {
      atomicAdd(msg_sum + i, s_sum[i]);
      atomicAdd(msg_ssq + i, s_ssq[i]);
    }
  }
}

// ------------------------- fold BN stats into per-channel scale/shift (a,b)
__global__ void finalize_stats(const float* __restrict__ sum,
                               const float* __restrict__ ssq,
                               const float* __restrict__ gamma,
                               const float* __restrict__ beta,
                               float* __restrict__ A, float* __restrict__ B,
                               float cnt, int C) {
  int c = blockIdx.x * blockDim.x + threadIdx.x;
  if (c < C) {
    float mean = sum[c] / cnt;
    float var = ssq[c] / cnt - mean * mean;  // biased variance, training mode
    float a = gamma[c] * rsqrtf(var + 1e-5f);
    A[c] = a;
    B[c] = beta[c] - mean * a;
  }
}

__global__ void node_stats_kernel(const float* __restrict__ h,
                                  float* __restrict__ sum,
                                  float* __restrict__ ssq, int N_) {
  __shared__ float s_s[NF], s_q[NF];
  for (int i = threadIdx.x; i < NF; i += blockDim.x) { s_s[i] = 0.f; s_q[i] = 0.f; }
  __syncthreads();
  float ps = 0.f, pq = 0.f;
  const int c = threadIdx.x & 63;
  const int rowStart = blockIdx.x * (blockDim.x >> 6) + (threadIdx.x >> 6);
  const int rowStride = gridDim.x * (blockDim.x >> 6);
  for (int r = rowStart; r < N_; r += rowStride) {
    float v = h[(size_t)r * NF + c];
    ps += v; pq += v * v;
  }
  atomicAdd(&s_s[c], ps);
  atomicAdd(&s_q[c], pq);
  __syncthreads();
  for (int i = threadIdx.x; i < NF; i += blockDim.x) {
    atomicAdd(sum + i, s_s[i]);
    atomicAdd(ssq + i, s_q[i]);
  }
}

__global__ void node_update_kernel(float* __restrict__ x,
                                   const float* __restrict__ h,
                                   const float* __restrict__ A,
                                   const float* __restrict__ B, int N_) {
  int idx = blockIdx.x * blockDim.x + threadIdx.x;
  const int total = N_ * NF;
  for (; idx < total; idx += gridDim.x * blockDim.x) {
    const int c = idx & 63;
    x[idx] = softplus_f(x[idx] + A[c] * h[idx] + B[c]);
  }
}

__global__ void pool_accum_kernel(const float* __restrict__ x,
                                  const int* __restrict__ gid,
                                  float* __restrict__ pooled,
                                  float* __restrict__ counts, int N_) {
  int idx = blockIdx.x * blockDim.x + threadIdx.x;
  const int total = N_ * NF;
  for (; idx < total; idx += gridDim.x * blockDim.x) {
    const int node = idx >> 6, c = idx & 63;
    const int g = gid[node];
    atomicAdd(pooled + (size_t)g * NF + c, x[idx]);
    if (c == 0) atomicAdd(counts + g, 1.0f);
  }
}

__global__ void pool_div_kernel(const float* __restrict__ pooled,
                                const float* __restrict__ counts,
                                float* __restrict__ out, int G_) {
  int idx = blockIdx.x * blockDim.x + threadIdx.x;
  if (idx < G_ * NF)
    out[idx] = pooled[idx] / fmaxf(counts[idx >> 6], 1.0f);
}

// --------------------------------------------------------------------- host
extern "C" void kernel_launch(void* const* d_in, const int* in_sizes, int n_in,
                              void* d_out, int out_size, void* d_ws,
                              size_t ws_size, hipStream_t stream) {
  const int* atypes    = (const int*)d_in[0];
  const float* dist    = (const float*)d_in[1];
  const int* src       = (const int*)d_in[2];
  const int* dst       = (const int*)d_in[3];
  const int* gid       = (const int*)d_in[4];
  // d_in[5] = n_graphs (device scalar); derive G from out_size instead
  const float* table   = (const float*)d_in[6];
  const float* W_embed = (const float*)d_in[7];
  const float* b_embed = (const float*)d_in[8];
  const float* W_src   = (const float*)d_in[9];
  const float* b_src   = (const float*)d_in[10];
  const float* W_dst   = (const float*)d_in[11];
  const float* b_dst   = (const float*)d_in[12];
  const float* W_edge  = (const float*)d_in[13];
  const float* b_edge  = (const float*)d_in[14];
  const float* g_msg   = (const float*)d_in[15];
  const float* beta_msg= (const float*)d_in[16];
  const float* g_bn    = (const float*)d_in[17];
  const float* beta_bn = (const float*)d_in[18];
  const int N_ = in_sizes[0];
  const int E_ = in_sizes[1];
  const int G_ = out_size / NF;

  float* ws = (float*)d_ws;
  size_t off = 0;
  float* x        = ws + off; off += (size_t)N_ * NF;
  float* h        = ws + off; off += (size_t)N_ * NF;
  float* hsrc     = ws + off; off += (size_t)N_ * NF2;
  float* hdst     = ws + off; off += (size_t)N_ * NF2;
  float* msg_sum  = ws + off; off += NF2;
  float* msg_ssq  = ws + off; off += NF2;
  float* msg_a    = ws + off; off += NF2;
  float* msg_b    = ws + off; off += NF2;
  float* hid_sum  = ws + off; off += NF;
  float* hid_ssq  = ws + off; off += NF;
  float* hid_a    = ws + off; off += NF;
  float* hid_b    = ws + off; off += NF;
  float* pooled   = ws + off; off += (size_t)G_ * NF;
  float* counts   = ws + off; off += G_;

  embed_kernel<<<1024, 256, 0, stream>>>(atypes, table, W_embed, b_embed, x, N_);

  for (int l = 0; l < LAYERS; ++l) {
    hipMemsetAsync(h, 0, (size_t)N_ * NF * sizeof(float), stream);
    hipMemsetAsync(msg_sum, 0, 2 * NF2 * sizeof(float), stream);  // sum+ssq contiguous
    hipMemsetAsync(hid_sum, 0, 2 * NF * sizeof(float), stream);

    node_proj_kernel<<<416, 256, 0, stream>>>(x, W_src + (size_t)l * NF * NF2,
                                              b_src + l * NF2, hsrc, N_);
    node_proj_kernel<<<416, 256, 0, stream>>>(x, W_dst + (size_t)l * NF * NF2,
                                              b_dst + l * NF2, hdst, N_);
    edge_kernel<1><<<1024, 256, 0, stream>>>(dist, src, dst,
        W_edge + (size_t)l * EFD * NF2, b_edge + l * NF2, hsrc, hdst,
        msg_sum, msg_ssq, nullptr, nullptr, nullptr, E_);
    finalize_stats<<<1, NF2, 0, stream>>>(msg_sum, msg_ssq, g_msg + l * NF2,
        beta_msg + l * NF2, msg_a, msg_b, (float)E_, NF2);
    edge_kernel<0><<<1024, 256, 0, stream>>>(dist, src, dst,
        W_edge + (size_t)l * EFD * NF2, b_edge + l * NF2, hsrc, hdst,
        nullptr, nullptr, msg_a, msg_b, h, E_);
    node_stats_kernel<<<512, 256, 0, stream>>>(h, hid_sum, hid_ssq, N_);
    finalize_stats<<<1, NF, 0, stream>>>(hid_sum, hid_ssq, g_bn + l * NF,
        beta_bn + l * NF, hid_a, hid_b, (float)N_, NF);
    node_update_kernel<<<1024, 256, 0, stream>>>(x, h, hid_a, hid_b, N_);
  }

  hipMemsetAsync(pooled, 0, ((size_t)G_ * NF + G_) * sizeof(float), stream);
  pool_accum_kernel<<<1024, 256, 0, stream>>>(x, gid, pooled, counts, N_);
  pool_div_kernel<<<(G_ * NF + 255) / 256, 256, 0, stream>>>(pooled, counts,
                                                             (float*)d_out, G_);
}